// GCNN_69114613730641
// MI455X (gfx1250) — compile-verified
//
#include <hip/hip_runtime.h>
#include <hip/hip_bf16.h>

#define F_IN  92
#define F_OUT 64

typedef __attribute__((ext_vector_type(2))) float v2f;
typedef __attribute__((ext_vector_type(8))) float v8f;

// Native CDNA5 fire-and-forget f32 atomic add (non-returning -> STOREcnt only,
// guaranteed GLOBAL_ATOMIC_ADD_F32 instead of a possible cmpswap retry loop).
__device__ __forceinline__ void gatomic_fadd(float* p, float v) {
    asm volatile("global_atomic_add_f32 %0, %1, off"
                 :
                 : "v"(p), "v"(v)
                 : "memory");
}

// ---------------------------------------------------------------------------
// deg[c] += 1 for every edge target (self-loops handled analytically later)
// ---------------------------------------------------------------------------
__global__ __launch_bounds__(256) void degree_kernel(const int* __restrict__ col,
                                                     float* __restrict__ deg, int E) {
    int e = blockIdx.x * blockDim.x + threadIdx.x;
    if (e < E) gatomic_fadd(&deg[col[e]], 1.0f);
}

// dis[i] = rsqrt(deg[i] + 1)   (+1 = self loop; deg >= 1 always)
__global__ __launch_bounds__(256) void dinv_kernel(const float* __restrict__ deg,
                                                   float* __restrict__ dis, int N) {
    int i = blockIdx.x * blockDim.x + threadIdx.x;
    if (i < N) dis[i] = rsqrtf(deg[i] + 1.0f);
}

// ---------------------------------------------------------------------------
// h = x @ W + b  via V_WMMA_F32_16X16X4_F32 (full fp32 precision)
// block = 128 threads = 4 waves; each wave owns one 16-wide N tile,
// all waves share one 16-row M tile of x staged in LDS.
// ---------------------------------------------------------------------------
__global__ __launch_bounds__(128) void gemm_wmma_kernel(const float* __restrict__ x,
                                                        const float* __restrict__ W,
                                                        const float* __restrict__ bias,
                                                        float* __restrict__ h, int N) {
    __shared__ float xs[16 * F_IN];

    const int m0  = blockIdx.x * 16;
    const int tid = threadIdx.x;

    // cooperative load of 16 x 92 tile of x into LDS
    for (int i = tid; i < 16 * F_IN; i += 128) {
        int row = m0 + i / F_IN;
        if (row >= N) row = N - 1;               // clamp (keeps EXEC full for WMMA)
        xs[i] = x[(long long)row * F_IN + (i % F_IN)];
    }
    __syncthreads();

    const int lane  = tid & 31;
    const int lhalf = lane >> 4;                 // 0 -> K pair {0,1}, 1 -> {2,3}
    const int l15   = lane & 15;
    const int n     = (tid >> 5) * 16 + l15;     // output column, per-wave N tile

    v8f acc = {};
    for (int kk = 0; kk < F_IN; kk += 4) {       // 23 fp32 WMMAs
        const int ka = kk + lhalf * 2;
        v2f a, bv;
        a[0]  = xs[l15 * F_IN + ka];             // A: 16x4 tile of x (row-major)
        a[1]  = xs[l15 * F_IN + ka + 1];
        bv[0] = W[(long long)ka * F_OUT + n];    // B: 4x16 tile of W (row-major)
        bv[1] = W[(long long)(ka + 1) * F_OUT + n];
        acc = __builtin_amdgcn_wmma_f32_16x16x4_f32(
            /*neg_a=*/false, a, /*neg_b=*/false, bv,
            /*c_mod=*/(short)0, acc, /*reuse_a=*/false, /*reuse_b=*/false);
    }

    const float bn   = bias[n];
    const int   mrow = m0 + lhalf * 8;           // C/D layout: VGPR r -> M = r + 8*lhalf
    for (int r = 0; r < 8; ++r) {
        const int row = mrow + r;
        if (row < N) h[(long long)row * F_OUT + n] = acc[r] + bn;
    }
}

// ---------------------------------------------------------------------------
// edge scatter: one wave per edge, each lane handles 2 consecutive features.
// out[col] += dis[row]*dis[col] * h[row]   (native f32 atomics, L2-resident)
// ---------------------------------------------------------------------------
__global__ __launch_bounds__(256) void scatter_kernel(const int* __restrict__ ei,
                                                      const float* __restrict__ h,
                                                      const float* __restrict__ dis,
                                                      float* __restrict__ out, int E) {
    long long t = (long long)blockIdx.x * blockDim.x + threadIdx.x;
    int e = (int)(t >> 5);
    if (e >= E) return;
    const int lane = threadIdx.x & 31;

    const int   r   = ei[e];                     // wave-uniform loads (broadcast)
    const int   c   = ei[E + e];
    const float nrm = dis[r] * dis[c];

    const float2 hv = *(const float2*)(h + (long long)r * F_OUT + lane * 2);
    float*       op = out + (long long)c * F_OUT + lane * 2;
    gatomic_fadd(op,     nrm * hv.x);
    gatomic_fadd(op + 1, nrm * hv.y);
}

// self-loop term: out[i] += (1/deg_with_loop[i]) * h[i]; no atomics needed
// (runs after scatter_kernel in stream order; unique element per thread)
__global__ __launch_bounds__(256) void selfloop_kernel(const float* __restrict__ h,
                                                       const float* __restrict__ dis,
                                                       float* __restrict__ out, int N) {
    long long t = (long long)blockIdx.x * blockDim.x + threadIdx.x;
    if (t >= (long long)N * F_OUT) return;
    const int   i = (int)(t >> 6);
    const float d = dis[i];
    out[t] += d * d * h[t];
}

// ---------------------------------------------------------------------------
extern "C" void kernel_launch(void* const* d_in, const int* in_sizes, int n_in,
                              void* d_out, int out_size, void* d_ws, size_t ws_size,
                              hipStream_t stream) {
    const float* x  = (const float*)d_in[0];   // [N, 92]
    const int*   ei = (const int*)d_in[1];     // [2, E] (jax default x32 -> int32)
    const float* W  = (const float*)d_in[2];   // [92, 64]
    const float* b  = (const float*)d_in[3];   // [64]
    float*       out = (float*)d_out;          // [N, 64]

    const int N = in_sizes[0] / F_IN;
    const int E = in_sizes[1] / 2;

    // workspace layout: h [N*64] | deg [N] | dis [N]  (~26.4 MB)
    float* h   = (float*)d_ws;
    float* deg = h + (size_t)N * F_OUT;
    float* dis = deg + N;

    hipMemsetAsync(out, 0, (size_t)out_size * sizeof(float), stream);
    hipMemsetAsync(deg, 0, (size_t)N * sizeof(float), stream);

    degree_kernel<<<(E + 255) / 256, 256, 0, stream>>>(ei + E, deg, E);
    dinv_kernel<<<(N + 255) / 256, 256, 0, stream>>>(deg, dis, N);

    gemm_wmma_kernel<<<(N + 15) / 16, 128, 0, stream>>>(x, W, b, h, N);

    const long long st = (long long)E * 32;
    scatter_kernel<<<(int)((st + 255) / 256), 256, 0, stream>>>(ei, h, dis, out, E);

    const long long lt = (long long)N * F_OUT;
    selfloop_kernel<<<(int)((lt + 255) / 256), 256, 0, stream>>>(h, dis, out, N);
}